// LocalBranch_11888469475440
// MI455X (gfx1250) — compile-verified
//
#include <hip/hip_runtime.h>
#include <hip/hip_bf16.h>

// ---------------------------------------------------------------------------
// Problem constants (from the reference)
// ---------------------------------------------------------------------------
#define BATCH 32
#define NPTS  16384
#define CHN   15        // 3 + CLS
#define CLSN  12
#define NA    96        // N_AGENT
#define NK    20        // N_KNN
#define NOUT  (NPTS + NA)   // 16480
#define RADIUSF 0.2f

typedef __attribute__((ext_vector_type(2))) float v2f;
typedef __attribute__((ext_vector_type(8))) float v8f;

#define V8ZERO {0.f,0.f,0.f,0.f,0.f,0.f,0.f,0.f}

// ---------------------------------------------------------------------------
// f32 WMMA 16x16x4 tile helper.
//   A: 16x4 (MxK) row-major with leading dim lda
//   B: 4x16 (KxN) row-major with leading dim ldb
// Fragment layout per CDNA5 ISA 7.12.2:
//   A: lanes 0-15 -> M=lane, K=0,1 ; lanes 16-31 -> M=lane-16, K=2,3
//   B: lanes 0-15 -> N=lane, K=0,1 ; lanes 16-31 -> N=lane-16, K=2,3
//   C/D: lane n=l&15, rows (l>>4)*8 + r (r = vgpr 0..7)
// ---------------------------------------------------------------------------
__device__ __forceinline__ v8f wmma4(const float* A, int lda,
                                     const float* B, int ldb, v8f c) {
  int l  = threadIdx.x & 31;
  int mn = l & 15;
  int kb = (l >> 4) << 1;   // 0 or 2
  v2f a, b;
  a.x = A[mn * lda + kb];
  a.y = A[mn * lda + kb + 1];
  b.x = B[kb * ldb + mn];
  b.y = B[(kb + 1) * ldb + mn];
  return __builtin_amdgcn_wmma_f32_16x16x4_f32(false, a, false, b,
                                               (short)0, c, false, false);
}

// ---------------------------------------------------------------------------
// K1: trans_dist[b][p] = sum_c trans_cord[b][c][p]^2
// ---------------------------------------------------------------------------
__global__ void k_dist(const float* __restrict__ trans, float* __restrict__ dist) {
  int i = blockIdx.x * blockDim.x + threadIdx.x;
  if (i >= BATCH * NPTS) return;
  int b = i / NPTS, p = i % NPTS;
  const float* base = trans + (size_t)b * CHN * NPTS + p;
  float s = 0.f;
#pragma unroll
  for (int c = 0; c < CHN; ++c) { float v = base[(size_t)c * NPTS]; s += v * v; }
  dist[i] = s;
}

// ---------------------------------------------------------------------------
// K2: per-batch top-96 by trans_dist (descending, ties -> lower index),
//     destructive on ws copy; then gather pcd_agent into ws.
// ---------------------------------------------------------------------------
__global__ void k_topk_agent(float* __restrict__ dist, const float* __restrict__ pcd,
                             int* __restrict__ idx_agent, float* __restrict__ pcd_agent) {
  __shared__ float sv[256];
  __shared__ int   si[256];
  __shared__ int   ssel[NA];
  int b = blockIdx.x, tid = threadIdx.x;
  float* d = dist + (size_t)b * NPTS;
  for (int j = 0; j < NA; ++j) {
    float bv = -__builtin_inff(); int bi = 0x7fffffff;
    for (int p = tid; p < NPTS; p += 256) {
      float v = d[p];
      if (v > bv || (v == bv && p < bi)) { bv = v; bi = p; }
    }
    sv[tid] = bv; si[tid] = bi;
    __syncthreads();
    for (int s = 128; s > 0; s >>= 1) {
      if (tid < s) {
        float ov = sv[tid + s]; int oi = si[tid + s];
        if (ov > sv[tid] || (ov == sv[tid] && oi < si[tid])) { sv[tid] = ov; si[tid] = oi; }
      }
      __syncthreads();
    }
    if (tid == 0) { ssel[j] = si[0]; d[si[0]] = -__builtin_inff(); }
    __syncthreads();
  }
  if (tid < NA) idx_agent[b * NA + tid] = ssel[tid];
  for (int t = tid; t < CHN * NA; t += 256) {
    int c = t / NA, j = t % NA;
    pcd_agent[((size_t)b * CHN + c) * NA + j] =
        pcd[(size_t)b * CHN * NPTS + (size_t)c * NPTS + ssel[j]];
  }
}

// ---------------------------------------------------------------------------
// K3: bulk float4 copy of the "passthrough" halves of both outputs.
//     pcd_local[:, :, :N] = pcd_coarse ; k_out[:, :, :N] = k_prev
// ---------------------------------------------------------------------------
__global__ void k_copy(const float4* __restrict__ pcd4, const float4* __restrict__ kp4,
                       float4* __restrict__ out4) {
  const int Q  = NPTS / 4;   // 4096
  const int QO = NOUT / 4;   // 4120 (16B aligned rows)
  long long i = (long long)blockIdx.x * blockDim.x + threadIdx.x;
  const long long totalP = (long long)BATCH * CHN * Q;
  const long long totalK = (long long)BATCH * 64 * Q;
  if (i < totalP) {
    long long row = i / Q, q = i % Q;
    out4[row * QO + q] = pcd4[row * Q + q];
  } else if (i < totalP + totalK) {
    long long j = i - totalP;
    long long row = j / Q, q = j % Q;
    out4[(long long)BATCH * CHN * QO + row * QO + q] = kp4[row * Q + q];
  }
}

// ---------------------------------------------------------------------------
// K3b: gathered agent columns of k_out.
// ---------------------------------------------------------------------------
__global__ void k_gather_k(const float* __restrict__ kp, const int* __restrict__ ia,
                           float* __restrict__ out) {
  int i = blockIdx.x * blockDim.x + threadIdx.x;
  if (i >= BATCH * 64 * NA) return;
  int b = i / (64 * NA);
  int r = i % (64 * NA);
  int c = r / NA, j = r % NA;
  size_t OUT_K = (size_t)BATCH * CHN * NOUT;
  out[OUT_K + ((size_t)b * 64 + c) * NOUT + NPTS + j] =
      kp[((size_t)b * 64 + c) * NPTS + ia[b * NA + j]];
}

// ---------------------------------------------------------------------------
// K4: KNN. One wave32 per agent. Each lane scans 512 interleaved points,
//     keeps a register-resident sorted top-20 (d2 asc, index tie-break),
//     then 32-way merge in LDS to the final 20 indices.
//     d2 = |a|^2 + |c|^2 - 2 a.c   (matches reference formula)
// ---------------------------------------------------------------------------
__global__ void __launch_bounds__(32) k_knn(const float* __restrict__ pcd,
                                            const float* __restrict__ pcd_agent,
                                            int* __restrict__ idx_knn) {
  int aid  = blockIdx.x;            // b*NA + j
  int b    = aid / NA;
  int j    = aid % NA;
  int lane = threadIdx.x;
  const float* cx = pcd + (size_t)b * CHN * NPTS;
  float ax = pcd_agent[((size_t)b * CHN + 0) * NA + j];
  float ay = pcd_agent[((size_t)b * CHN + 1) * NA + j];
  float az = pcd_agent[((size_t)b * CHN + 2) * NA + j];
  float qq = ax * ax + ay * ay + az * az;

  float bd[NK]; int bix[NK];
#pragma unroll
  for (int s = 0; s < NK; ++s) { bd[s] = __builtin_inff(); bix[s] = 0x7fffffff; }

  for (int p0 = 0; p0 < NPTS; p0 += 32) {
    int p = p0 + lane;
    float x = cx[p], y = cx[NPTS + p], z = cx[2 * NPTS + p];
    float rr = x * x + y * y + z * z;
    float qr = ax * x + ay * y + az * z;
    float d2 = qq + rr - 2.f * qr;
    bool ins = (d2 < bd[NK - 1]) || (d2 == bd[NK - 1] && p < bix[NK - 1]);
    if (ins) {
      float cd = d2; int ci = p;
#pragma unroll
      for (int s = 0; s < NK; ++s) {
        bool sw = (cd < bd[s]) || (cd == bd[s] && ci < bix[s]);
        if (sw) { float td = bd[s]; int ti = bix[s]; bd[s] = cd; bix[s] = ci; cd = td; ci = ti; }
      }
    }
  }

  __shared__ float md[32 * NK];
  __shared__ int   mi[32 * NK];
  __shared__ int   heads[32];
#pragma unroll
  for (int s = 0; s < NK; ++s) { md[lane * NK + s] = bd[s]; mi[lane * NK + s] = bix[s]; }
  heads[lane] = 0;
  __syncthreads();
  if (lane == 0) {
    int* outk = idx_knn + (size_t)aid * NK;
    for (int r = 0; r < NK; ++r) {
      float best = __builtin_inff(); int bi2 = 0x7fffffff; int bl = 0;
      for (int l2 = 0; l2 < 32; ++l2) {
        int h = heads[l2];
        float v = md[l2 * NK + h]; int iv = mi[l2 * NK + h];
        if (v < best || (v == best && iv < bi2)) { best = v; bi2 = iv; bl = l2; }
      }
      heads[bl]++;
      outk[r] = bi2;
    }
  }
}

// ---------------------------------------------------------------------------
// K5: per-agent patch MLPs via WMMA f32 16x16x4.
//   layer1: (32 x Kin) @ (Kin x 16pts)  -> relu
//   layer2: (64 x 32)  @ (32 x 16pts)
//   feat   = relu(max over first 20 pts)
// ---------------------------------------------------------------------------
__device__ __forceinline__ void feat_branch(
    int lane, const float* __restrict__ src, int nch, int Kp,
    const float* __restrict__ W1, const float* __restrict__ b1,
    const float* __restrict__ W2, const float* __restrict__ b2,
    const int* sIdx,
    float (*sPatch)[32], float (*sW1)[12], float (*sW2)[32],
    float (*sH1)[32], float (*sH2)[32],
    float* __restrict__ featOut) {
#pragma unroll
  for (int r = 0; r < 12; ++r) sPatch[r][lane] = 0.f;
  __syncthreads();
  if (lane < NK) {
    int p  = sIdx[lane];
    int p0 = sIdx[0];
    for (int c2 = 0; c2 < nch; ++c2)
      sPatch[c2][lane] = src[(size_t)c2 * NPTS + p] - src[(size_t)c2 * NPTS + p0];
  }
  {
    int o = lane;   // 32 weight rows, one per lane
    for (int c2 = 0; c2 < 12; ++c2)
      sW1[o][c2] = (c2 < nch) ? W1[o * nch + c2] : 0.f;
  }
  __syncthreads();

  int nks1 = Kp >> 2;
  int nb = lane & 15, rb = (lane >> 4) * 8;
  for (int mt = 0; mt < 2; ++mt) {
    v8f acc = V8ZERO;
    for (int ks = 0; ks < nks1; ++ks)
      acc = wmma4(&sW1[mt * 16][ks * 4], 12, &sPatch[ks * 4][0], 32, acc);
#pragma unroll
    for (int r = 0; r < 8; ++r) {
      int o = mt * 16 + rb + r;
      float v = acc[r] + b1[o];
      sH1[o][nb] = v > 0.f ? v : 0.f;
    }
  }
  __syncthreads();

  for (int t = lane; t < 64 * 32; t += 32) sW2[t >> 5][t & 31] = W2[t];
  __syncthreads();

  for (int mt = 0; mt < 4; ++mt) {
    v8f acc = V8ZERO;
#pragma unroll
    for (int ks = 0; ks < 8; ++ks)
      acc = wmma4(&sW2[mt * 16][ks * 4], 32, &sH1[ks * 4][0], 32, acc);
#pragma unroll
    for (int r = 0; r < 8; ++r) {
      int o = mt * 16 + rb + r;
      sH2[o][nb] = acc[r] + b2[o];
    }
  }
  __syncthreads();

#pragma unroll
  for (int t = 0; t < 2; ++t) {
    int o = lane + t * 32;
    float f = -__builtin_inff();
    for (int jj = 0; jj < NK; ++jj) f = fmaxf(f, sH2[o][jj]);
    featOut[o] = fmaxf(f, 0.f);   // relu(max(.))
  }
  __syncthreads();
}

__global__ void __launch_bounds__(32) k_feat(
    const float* __restrict__ pcd, const int* __restrict__ idx_knn,
    const float* __restrict__ cW1, const float* __restrict__ cb1,
    const float* __restrict__ cW2, const float* __restrict__ cb2,
    const float* __restrict__ lW1, const float* __restrict__ lb1,
    const float* __restrict__ lW2, const float* __restrict__ lb2,
    float* __restrict__ featC, float* __restrict__ featL) {
  __shared__ float sPatch[12][32];
  __shared__ float sW1[32][12];
  __shared__ float sW2[64][32];
  __shared__ float sH1[32][32];
  __shared__ float sH2[64][32];
  __shared__ int   sIdx[NK];

  int aid  = blockIdx.x;
  int b    = aid / NA;
  int lane = threadIdx.x;
  if (lane < NK) sIdx[lane] = idx_knn[(size_t)aid * NK + lane];
  __syncthreads();

  const float* coords = pcd + (size_t)b * CHN * NPTS;
  const float* labs   = coords + (size_t)3 * NPTS;
  feat_branch(lane, coords, 3, 4,  cW1, cb1, cW2, cb2, sIdx,
              sPatch, sW1, sW2, sH1, sH2, featC + (size_t)aid * 64);
  feat_branch(lane, labs,  12, 12, lW1, lb1, lW2, lb2, sIdx,
              sPatch, sW1, sW2, sH1, sH2, featL + (size_t)aid * 64);
}

// ---------------------------------------------------------------------------
// K6: head MLPs (64->128->64->{3,12}) via WMMA over 16-agent tiles,
//     tanh * 0.2, add pcd_agent, write agent columns of pcd_local.
// ---------------------------------------------------------------------------
__device__ __forceinline__ void head_branch(
    int lane, int b, int a0,
    const float* __restrict__ feat,
    const float* __restrict__ W1, const float* __restrict__ b1,
    const float* __restrict__ W2, const float* __restrict__ b2,
    const float* __restrict__ W3, const float* __restrict__ b3,
    int nout, int chan_base,
    const float* __restrict__ pcd_agent, float* __restrict__ outp,
    float (*sF)[16], float (*sH1)[16], float (*sH2)[16],
    float (*sW3)[64], float* sWbuf) {
  for (int t = lane; t < 64 * 16; t += 32) {
    int k = t >> 4, n2 = t & 15;
    sF[k][n2] = feat[(size_t)(b * NA + a0 + n2) * 64 + k];
  }
  for (int t = lane; t < 8192; t += 32) sWbuf[t] = W1[t];   // 128x64
  __syncthreads();

  int nb = lane & 15, rb = (lane >> 4) * 8;
  for (int mt = 0; mt < 8; ++mt) {
    v8f acc = V8ZERO;
#pragma unroll
    for (int ks = 0; ks < 16; ++ks)
      acc = wmma4(&sWbuf[(mt * 16) * 64 + ks * 4], 64, &sF[ks * 4][0], 16, acc);
#pragma unroll
    for (int r = 0; r < 8; ++r) {
      int o = mt * 16 + rb + r;
      float v = acc[r] + b1[o];
      sH1[o][nb] = v > 0.f ? v : 0.f;
    }
  }
  __syncthreads();

  for (int t = lane; t < 8192; t += 32) sWbuf[t] = W2[t];   // 64x128
  __syncthreads();

  for (int mt = 0; mt < 4; ++mt) {
    v8f acc = V8ZERO;
#pragma unroll
    for (int ks = 0; ks < 32; ++ks)
      acc = wmma4(&sWbuf[(mt * 16) * 128 + ks * 4], 128, &sH1[ks * 4][0], 16, acc);
#pragma unroll
    for (int r = 0; r < 8; ++r) {
      int o = mt * 16 + rb + r;
      float v = acc[r] + b2[o];
      sH2[o][nb] = v > 0.f ? v : 0.f;
    }
  }

  for (int t = lane; t < 16 * 64; t += 32) {         // zero-padded W3 (nout x 64)
    int r2 = t >> 6, c2 = t & 63;
    sW3[r2][c2] = (r2 < nout) ? W3[r2 * 64 + c2] : 0.f;
  }
  __syncthreads();

  v8f acc = V8ZERO;
#pragma unroll
  for (int ks = 0; ks < 16; ++ks)
    acc = wmma4(&sW3[0][ks * 4], 64, &sH2[ks * 4][0], 16, acc);

#pragma unroll
  for (int r = 0; r < 8; ++r) {
    int row = rb + r;
    if (row < nout) {
      int c2 = chan_base + row;
      int ag = a0 + nb;
      float v = tanhf(acc[r] + b3[row]) * RADIUSF;
      v += pcd_agent[((size_t)b * CHN + c2) * NA + ag];
      outp[(size_t)b * CHN * NOUT + (size_t)c2 * NOUT + NPTS + ag] = v;
    }
  }
  __syncthreads();
}

__global__ void __launch_bounds__(32) k_head(
    const float* __restrict__ featC, const float* __restrict__ featL,
    const float* __restrict__ cmW1, const float* __restrict__ cmb1,
    const float* __restrict__ cmW2, const float* __restrict__ cmb2,
    const float* __restrict__ cmW3, const float* __restrict__ cmb3,
    const float* __restrict__ dW1,  const float* __restrict__ db1,
    const float* __restrict__ dW2,  const float* __restrict__ db2,
    const float* __restrict__ dW3,  const float* __restrict__ db3,
    const float* __restrict__ pcd_agent, float* __restrict__ out) {
  __shared__ float sF[64][16];
  __shared__ float sH1[128][16];
  __shared__ float sH2[64][16];
  __shared__ float sW3[16][64];
  __shared__ float sWbuf[8192];
  int lane = threadIdx.x;
  int tile = blockIdx.x;                // 0 .. BATCH*6-1
  int b = tile / 6, a0 = (tile % 6) * 16;
  head_branch(lane, b, a0, featC, cmW1, cmb1, cmW2, cmb2, cmW3, cmb3,
              3, 0, pcd_agent, out, sF, sH1, sH2, sW3, sWbuf);
  head_branch(lane, b, a0, featL, dW1, db1, dW2, db2, dW3, db3,
              CLSN, 3, pcd_agent, out, sF, sH1, sH2, sW3, sWbuf);
}

// ---------------------------------------------------------------------------
// Host launcher
// ---------------------------------------------------------------------------
extern "C" void kernel_launch(void* const* d_in, const int* in_sizes, int n_in,
                              void* d_out, int out_size, void* d_ws, size_t ws_size,
                              hipStream_t stream) {
  const float* pcd   = (const float*)d_in[0];
  const float* trans = (const float*)d_in[1];
  const float* kprev = (const float*)d_in[2];
  const float* cW1 = (const float*)d_in[3];
  const float* cb1 = (const float*)d_in[4];
  const float* cW2 = (const float*)d_in[5];
  const float* cb2 = (const float*)d_in[6];
  const float* lW1 = (const float*)d_in[7];
  const float* lb1 = (const float*)d_in[8];
  const float* lW2 = (const float*)d_in[9];
  const float* lb2 = (const float*)d_in[10];
  const float* cmW1 = (const float*)d_in[11];
  const float* cmb1 = (const float*)d_in[12];
  const float* cmW2 = (const float*)d_in[13];
  const float* cmb2 = (const float*)d_in[14];
  const float* cmW3 = (const float*)d_in[15];
  const float* cmb3 = (const float*)d_in[16];
  const float* dW1 = (const float*)d_in[17];
  const float* db1 = (const float*)d_in[18];
  const float* dW2 = (const float*)d_in[19];
  const float* db2 = (const float*)d_in[20];
  const float* dW3 = (const float*)d_in[21];
  const float* db3 = (const float*)d_in[22];
  float* out = (float*)d_out;

  // workspace layout (~4.2 MB total, all 16B aligned)
  char* ws = (char*)d_ws;
  float* ws_dist = (float*)ws;  ws += sizeof(float) * BATCH * NPTS;
  int*   ws_ia   = (int*)ws;    ws += sizeof(int)   * BATCH * NA;
  float* ws_pa   = (float*)ws;  ws += sizeof(float) * BATCH * CHN * NA;
  int*   ws_knn  = (int*)ws;    ws += sizeof(int)   * BATCH * NA * NK;
  float* ws_fc   = (float*)ws;  ws += sizeof(float) * BATCH * NA * 64;
  float* ws_fl   = (float*)ws;

  k_dist<<<(BATCH * NPTS + 255) / 256, 256, 0, stream>>>(trans, ws_dist);
  k_topk_agent<<<BATCH, 256, 0, stream>>>(ws_dist, pcd, ws_ia, ws_pa);

  long long total4 = (long long)BATCH * (CHN + 64) * (NPTS / 4);
  k_copy<<<(int)((total4 + 255) / 256), 256, 0, stream>>>(
      (const float4*)pcd, (const float4*)kprev, (float4*)out);
  k_gather_k<<<(BATCH * 64 * NA + 255) / 256, 256, 0, stream>>>(kprev, ws_ia, out);

  k_knn<<<BATCH * NA, 32, 0, stream>>>(pcd, ws_pa, ws_knn);
  k_feat<<<BATCH * NA, 32, 0, stream>>>(pcd, ws_knn,
                                        cW1, cb1, cW2, cb2,
                                        lW1, lb1, lW2, lb2,
                                        ws_fc, ws_fl);
  k_head<<<BATCH * 6, 32, 0, stream>>>(ws_fc, ws_fl,
                                       cmW1, cmb1, cmW2, cmb2, cmW3, cmb3,
                                       dW1, db1, dW2, db2, dW3, db3,
                                       ws_pa, out);
}